// fcn_51505247813778
// MI455X (gfx1250) — compile-verified
//
#include <hip/hip_runtime.h>

// ---------------------------------------------------------------------------
// Spherical FCN forward for MI455X (gfx1250, wave32, WMMA).
// GEMMs: V_WMMA_F32_16X16X4_F32, 64x64 block tile, 4 waves, each wave 16x64
// (4 accumulators), double-buffered LDS, 1 barrier per K-chunk.
// ---------------------------------------------------------------------------

typedef float v2f __attribute__((ext_vector_type(2)));
typedef float v8f __attribute__((ext_vector_type(8)));

#define EPS_BN 1e-5f

// ---------------------------------------------------------------------------
// gather7: G[r, j*f + c] = x[neigh[r*7+j]*f + c], zero-padded to npad x Kpad
// ---------------------------------------------------------------------------
__global__ __launch_bounds__(256) void gather7(
    const float* __restrict__ x, const int* __restrict__ neigh,
    float* __restrict__ G, int n, int f, int npad, int Kpad, long total) {
  long idx = (long)blockIdx.x * 256 + threadIdx.x;
  if (idx >= total) return;
  int K = 7 * f;
  int r = (int)(idx / Kpad);
  int k = (int)(idx - (long)r * Kpad);
  float v = 0.f;
  if (r < n && k < K) {
    int j = k / f;
    int c = k - j * f;
    v = x[(size_t)neigh[r * 7 + j] * f + c];
  }
  G[idx] = v;
}

// ---------------------------------------------------------------------------
// WMMA fp32 GEMM: Y[n,fo] = A[npad,Kpad] @ W[fo,K]^T + bias
// Block: 128 threads = 4 waves. Block tile 64(M) x 64(N); wave tile 16x64.
// Double-buffered LDS, software-pipelined global loads, 1 barrier / chunk.
// Fragment layouts per CDNA5 ISA 7.12.2:
//   A 16x4 : lane<16 -> {K0,K1}, lane>=16 -> {K2,K3}, M = lane&15
//   B 4x16 : lane<16 -> {K0,K1}, lane>=16 -> {K2,K3}, N = lane&15
//   C/D    : vgpr i : lane<16 -> M=i, lane>=16 -> M=i+8, N = lane&15
// ---------------------------------------------------------------------------
__global__ __launch_bounds__(128) void gemm_wmma_f32(
    const float* __restrict__ A, const float* __restrict__ W,
    const float* __restrict__ bias, float* __restrict__ Y,
    int n, int K, int Kpad, int fo) {
  __shared__ float As[2][64][36];
  __shared__ float Ws[2][64][36];

  const int tid  = threadIdx.x;
  const int wave = tid >> 5;
  const int lane = tid & 31;
  const int half = lane >> 4;
  const int l    = lane & 15;
  const int m0   = blockIdx.y * 64;
  const int nb0  = blockIdx.x * 64;
  const int arow = tid >> 3;        // 0..15
  const int ac4  = (tid & 7) * 4;   // 0,4,...,28

  const bool Kvec   = ((K & 3) == 0);
  const int nchunks = Kpad / 32;

  float4 ra[4], rw[4];

  auto g_load = [&](int ch) {
    const int k0 = ch * 32;
#pragma unroll
    for (int j = 0; j < 4; ++j)
      ra[j] = *(const float4*)(A + (size_t)(m0 + arow + j * 16) * Kpad + k0 + ac4);
    if (Kvec) {
#pragma unroll
      for (int j = 0; j < 4; ++j) {
        int gr = nb0 + arow + j * 16;
        float4 vw = make_float4(0.f, 0.f, 0.f, 0.f);
        if (gr < fo && (k0 + ac4 + 4) <= K)
          vw = *(const float4*)(W + (size_t)gr * K + k0 + ac4);
        rw[j] = vw;
      }
    } else {
#pragma unroll
      for (int j = 0; j < 4; ++j) {
        int gr = nb0 + arow + j * 16;
        float4 vw;
        float* pw = (float*)&vw;
#pragma unroll
        for (int u = 0; u < 4; ++u) {
          int k = k0 + ac4 + u;
          pw[u] = (gr < fo && k < K) ? W[(size_t)gr * K + k] : 0.f;
        }
        rw[j] = vw;
      }
    }
  };
  auto s_store = [&](int buf) {
#pragma unroll
    for (int j = 0; j < 4; ++j) {
      *(float4*)&As[buf][arow + j * 16][ac4] = ra[j];
      *(float4*)&Ws[buf][arow + j * 16][ac4] = rw[j];
    }
  };

  v8f acc[4] = {};

  g_load(0);
  s_store(0);
  __syncthreads();

  for (int ch = 0; ch < nchunks; ++ch) {
    const int buf = ch & 1;
    if (ch + 1 < nchunks) g_load(ch + 1);   // overlap with WMMA below
#pragma unroll
    for (int kk = 0; kk < 8; ++kk) {
      const int kb = kk * 4 + half * 2;
      v2f a = *(const v2f*)(&As[buf][wave * 16 + l][kb]);
#pragma unroll
      for (int t = 0; t < 4; ++t) {
        v2f b = *(const v2f*)(&Ws[buf][t * 16 + l][kb]);
        acc[t] = __builtin_amdgcn_wmma_f32_16x16x4_f32(
            false, a, false, b, (short)0, acc[t], false, false);
      }
    }
    if (ch + 1 < nchunks) {
      s_store(buf ^ 1);   // other buffer: safe, last read before prev barrier
      __syncthreads();
    }
  }

#pragma unroll
  for (int t = 0; t < 4; ++t) {
    const int col = nb0 + t * 16 + l;
    const float bv = (col < fo) ? bias[col] : 0.f;
#pragma unroll
    for (int i = 0; i < 8; ++i) {
      int row = m0 + wave * 16 + i + half * 8;
      if (row < n && col < fo) Y[(size_t)row * fo + col] = acc[t][i] + bv;
    }
  }
}

// ---------------------------------------------------------------------------
// BN stats: coalesced row-slab accumulation + atomic reduction.
// Requires fo to be a power of two (64..1024 here). P = columns per thread.
// ---------------------------------------------------------------------------
__global__ __launch_bounds__(256) void bn_stats(
    const float* __restrict__ Y, long total, int fo, long slab,
    float* __restrict__ ssum, float* __restrict__ sqsum) {
  const int P = (fo > 256) ? (fo >> 8) : 1;   // 1,1,1,2,4
  long base = (long)blockIdx.x * slab;
  long end  = base + slab;
  if (end > total) end = total;
  float s[4] = {0.f, 0.f, 0.f, 0.f};
  float q[4] = {0.f, 0.f, 0.f, 0.f};
  int i = 0;
  for (long e = base + threadIdx.x; e < end; e += 256, ++i) {
    float v = Y[e];
    int p = i & (P - 1);
    s[p] += v;
    q[p] += v * v;
  }
#pragma unroll
  for (int p = 0; p < 4; ++p) {
    if (p < P) {
      int c = (P == 1) ? (threadIdx.x & (fo - 1)) : (threadIdx.x + (p << 8));
      atomicAdd(&ssum[c], s[p]);
      atomicAdd(&sqsum[c], q[p]);
    }
  }
}

__global__ __launch_bounds__(256) void bn_relu(
    const float* __restrict__ Y, const float* __restrict__ ssum,
    const float* __restrict__ sqsum, const float* __restrict__ g,
    const float* __restrict__ be, float* __restrict__ out,
    long total, int fo, float inv_n) {
  long idx = (long)blockIdx.x * 256 + threadIdx.x;
  if (idx >= total) return;
  int c = (int)(idx % fo);
  float m  = ssum[c] * inv_n;
  float vv = sqsum[c] * inv_n - m * m;
  float v = (Y[idx] - m) * rsqrtf(vv + EPS_BN) * g[c] + be[c];
  out[idx] = v > 0.f ? v : 0.f;
}

// ---------------------------------------------------------------------------
// pool7: replicates torch raw view(n_new, f, 7).mean(2)
// ---------------------------------------------------------------------------
__global__ __launch_bounds__(256) void pool7(
    const float* __restrict__ x, const int* __restrict__ neigh,
    float* __restrict__ out, int n_new, int f) {
  long idx = (long)blockIdx.x * 256 + threadIdx.x;
  if (idx >= (long)n_new * f) return;
  int i = (int)(idx / f);
  int c = (int)(idx - (long)i * f);
  long tbase = (long)i * 7 * f + (long)c * 7;
  float s = 0.f;
#pragma unroll
  for (int j = 0; j < 7; ++j) {
    long t = tbase + j;
    int r  = (int)(t / f);
    int cc = (int)(t - (long)r * f);
    s += x[(size_t)neigh[r] * f + cc];
  }
  out[idx] = s * (1.f / 7.f);
}

// ---------------------------------------------------------------------------
// upsample2: in-place on strided buffer; appends rows n..4n-7.
// Replicates raw view(3n-6, f, 2).mean(2). New rows read only rows < n.
// ---------------------------------------------------------------------------
__global__ __launch_bounds__(256) void upsample2(
    float* __restrict__ buf, const int* __restrict__ up, int n, int f, int ld) {
  int nn = 3 * n - 6;
  long idx = (long)blockIdx.x * 256 + threadIdx.x;
  if (idx >= (long)nn * f) return;
  int i = (int)(idx / f);
  int c = (int)(idx - (long)i * f);
  float s = 0.f;
#pragma unroll
  for (int j = 0; j < 2; ++j) {
    long t = (long)i * 2 * f + (long)c * 2 + j;
    int r  = (int)(t / f);
    int cc = (int)(t - (long)r * f);
    s += buf[(size_t)up[r] * ld + cc];
  }
  buf[(size_t)(n + i) * ld + c] = 0.5f * s;
}

__global__ __launch_bounds__(256) void copy_strided(
    const float* __restrict__ src, float* __restrict__ dst,
    int n, int f, int ld) {
  long idx = (long)blockIdx.x * 256 + threadIdx.x;
  if (idx >= (long)n * f) return;
  int r = (int)(idx / f);
  int c = (int)(idx - (long)r * f);
  dst[(size_t)r * ld + c] = src[idx];
}

__global__ __launch_bounds__(256) void zero_fill(float* __restrict__ p, long total) {
  long idx = (long)blockIdx.x * 256 + threadIdx.x;
  if (idx < total) p[idx] = 0.f;
}

// ---------------------------------------------------------------------------
// upconv final gather: y viewed as (n*7, 36); top rows + mean-of-2 down rows
// ---------------------------------------------------------------------------
__global__ __launch_bounds__(256) void upconv_out(
    const float* __restrict__ Y, const int* __restrict__ top,
    const int* __restrict__ down, float* __restrict__ out) {
  const int OC = 36;
  long idx = (long)blockIdx.x * 256 + threadIdx.x;
  if (idx >= 40962L * OC) return;
  int i = (int)(idx / OC);
  int c = (int)(idx - (long)i * OC);
  if (i < 10242) {
    out[idx] = Y[(size_t)top[i] * OC + c];
  } else {
    int i2 = i - 10242;
    float s = 0.f;
#pragma unroll
    for (int j = 0; j < 2; ++j) {
      long t = (long)i2 * 2 * OC + (long)c * 2 + j;
      int r  = (int)(t / OC);
      int cc = (int)(t - (long)r * OC);
      s += Y[(size_t)down[r] * OC + cc];
    }
    out[idx] = 0.5f * s;
  }
}

// ---------------------------------------------------------------------------
// Host orchestration
// ---------------------------------------------------------------------------
static inline int cdiv_l(long a, long b) { return (int)((a + b - 1) / b); }
static inline int rnd64(int n) { return ((n + 63) / 64) * 64; }

extern "C" void kernel_launch(void* const* d_in, const int* in_sizes, int n_in,
                              void* d_out, int out_size, void* d_ws, size_t ws_size,
                              hipStream_t stream) {
  (void)in_sizes; (void)n_in; (void)out_size; (void)ws_size;

  // --- inputs in setup_inputs() dict insertion order ---
  const float* x_in = (const float*)d_in[0];
  const float *W[10], *bb[10], *gg[10], *be[10];
  for (int i = 0; i < 10; ++i) {
    W[i]  = (const float*)d_in[1 + 4 * i];
    bb[i] = (const float*)d_in[2 + 4 * i];
    gg[i] = (const float*)d_in[3 + 4 * i];
    be[i] = (const float*)d_in[4 + 4 * i];
  }
  const float* Wup = (const float*)d_in[41];
  const float* bup = (const float*)d_in[42];
  const int* neigh[5];
  for (int i = 0; i < 5; ++i) neigh[i] = (const int*)d_in[43 + i];
  const int* up162  = (const int*)d_in[48];
  const int* up642  = (const int*)d_in[49];
  const int* up2562 = (const int*)d_in[50];
  const int* topi   = (const int*)d_in[51];
  const int* downi  = (const int*)d_in[52];

  // --- workspace layout (floats), 64-element aligned ---
  float* ws = (float*)d_ws;
  size_t off = 0;
  auto alloc = [&](size_t nfl) {
    float* p = ws + off;
    off += (nfl + 63) & ~(size_t)63;
    return p;
  };
  float* G    = alloc((size_t)41024 * 448);   // padded gather matrix (max: layer 1)
  float* CONC = alloc((size_t)10304 * 1792);  // padded concat [x5|x4|x3]
  float* Yb   = alloc((size_t)40962 * 64);    // GEMM output (max over layers)
  float* actA = alloc((size_t)40962 * 64);
  float* actB = alloc((size_t)40962 * 64);
  float* ssum = alloc(1024);
  float* sqsum = alloc(1024);                 // contiguous with ssum

  auto run_gather = [&](const float* xx, const int* ng, int n, int f) {
    int K = 7 * f, Kpad = ((K + 31) / 32) * 32, npad = rnd64(n);
    long total = (long)npad * Kpad;
    gather7<<<cdiv_l(total, 256), 256, 0, stream>>>(xx, ng, G, n, f, npad, Kpad, total);
  };
  auto run_gemm = [&](const float* A, const float* Wm, const float* bm, float* Y,
                      int n, int K, int fo) {
    int npad = rnd64(n);
    int Kpad = ((K + 31) / 32) * 32;
    dim3 grid((fo + 63) / 64, npad / 64);
    gemm_wmma_f32<<<grid, 128, 0, stream>>>(A, Wm, bm, Y, n, K, Kpad, fo);
  };
  auto run_bn = [&](int li, float* Y, float* outp, int n, int fo) {
    long total = (long)n * fo;
    zero_fill<<<cdiv_l(2048, 256), 256, 0, stream>>>(ssum, 2048);
    long slab = 256L * fo;  // 256 rows per block, coalesced
    bn_stats<<<cdiv_l(n, 256), 256, 0, stream>>>(Y, total, fo, slab, ssum, sqsum);
    bn_relu<<<cdiv_l(total, 256), 256, 0, stream>>>(
        Y, ssum, sqsum, gg[li], be[li], outp, total, fo, 1.f / (float)n);
  };
  auto cbr = [&](int li, const float* inx, float* outp, const int* ng,
                 int n, int fi, int fo) {
    run_gather(inx, ng, n, fi);
    run_gemm(G, W[li], bb[li], Yb, n, 7 * fi, fo);
    run_bn(li, Yb, outp, n, fo);
  };
  auto run_pool = [&](const float* xx, const int* ng, float* outp, int n_new, int f) {
    pool7<<<cdiv_l((long)n_new * f, 256), 256, 0, stream>>>(xx, ng, outp, n_new, f);
  };
  auto run_copy = [&](const float* src, float* dst, int n, int f) {
    copy_strided<<<cdiv_l((long)n * f, 256), 256, 0, stream>>>(src, dst, n, f, 1792);
  };
  auto run_up = [&](float* base, const int* up, int n, int f) {
    upsample2<<<cdiv_l((long)(3 * n - 6) * f, 256), 256, 0, stream>>>(base, up, n, f, 1792);
  };

  // ---- encoder ----
  cbr(0, x_in, actA, neigh[0], 40962, 3, 64);
  cbr(1, actA, actB, neigh[0], 40962, 64, 64);           // x1
  run_pool(actB, neigh[0], actA, 10242, 64);
  cbr(2, actA, actB, neigh[1], 10242, 64, 128);
  cbr(3, actB, actA, neigh[1], 10242, 128, 128);         // x2
  run_pool(actA, neigh[1], actB, 2562, 128);
  cbr(4, actB, actA, neigh[2], 2562, 128, 256);
  cbr(5, actA, actB, neigh[2], 2562, 256, 256);          // x3
  run_copy(actB, CONC + 1536, 2562, 256);
  run_pool(actB, neigh[2], actA, 642, 256);
  cbr(6, actA, actB, neigh[3], 642, 256, 512);
  cbr(7, actB, actA, neigh[3], 642, 512, 512);           // x4
  run_copy(actA, CONC + 1024, 642, 512);
  run_pool(actA, neigh[3], actB, 162, 512);
  cbr(8, actB, actA, neigh[4], 162, 512, 1024);
  cbr(9, actA, actB, neigh[4], 162, 1024, 1024);         // x5
  run_copy(actB, CONC + 0, 162, 1024);

  // ---- upsample chains in place inside CONC (ld = 1792) ----
  run_up(CONC + 0,    up162,  162,  1024);
  run_up(CONC + 0,    up642,  642,  1024);
  run_up(CONC + 0,    up2562, 2562, 1024);
  run_up(CONC + 1024, up642,  642,  512);
  run_up(CONC + 1024, up2562, 2562, 512);
  run_up(CONC + 1536, up2562, 2562, 256);

  // zero the pad rows of CONC (rows 10242..10303)
  zero_fill<<<cdiv_l(62L * 1792, 256), 256, 0, stream>>>(
      CONC + (size_t)10242 * 1792, 62L * 1792);

  // ---- upconv GEMM (10242 x 1792 -> 252) + final gather ----
  run_gemm(CONC, Wup, bup, Yb, 10242, 1792, 252);
  upconv_out<<<cdiv_l(40962L * 36, 256), 256, 0, stream>>>(
      Yb, topi, downi, (float*)d_out);
}